// PointPillar_78924319031400
// MI455X (gfx1250) — compile-verified
//
#include <hip/hip_runtime.h>
#include <stdint.h>

#define PI_F 3.14159265358979323846f
#define G_GT 20
#define NCLS 3

// ---- anchor-generator constants (match reference RANGES / SIZES / ROTATIONS / THRESH) ----
__constant__ float c_rng[NCLS][6] = {
    {0.0f, -39.68f, -0.60f, 69.12f, 39.68f, -0.60f},
    {0.0f, -39.68f, -0.60f, 69.12f, 39.68f, -0.60f},
    {0.0f, -39.68f, -1.78f, 69.12f, 39.68f, -1.78f}};
__constant__ float c_sz[NCLS][3] = {
    {0.6f, 0.8f, 1.73f}, {0.6f, 1.76f, 1.73f}, {1.6f, 3.9f, 1.56f}};
__constant__ float c_rot[2] = {0.0f, 1.57f};
__constant__ float c_pos[NCLS] = {0.50f, 0.50f, 0.60f};
__constant__ float c_neg[NCLS] = {0.35f, 0.35f, 0.45f};   // neg == min thr in all classes

// ---------------- CDNA5 async global->LDS staging (ASYNCcnt path) ----------------
__device__ __forceinline__ void async_ld_b128(uint32_t lds_addr, const void* gaddr) {
#if defined(__gfx1250__)
  asm volatile("global_load_async_to_lds_b128 %0, %1, off"
               :: "v"(lds_addr), "v"((unsigned long long)(uintptr_t)gaddr)
               : "memory");
#else
  for (int k = 0; k < 4; ++k)
    ((volatile uint32_t*)(uintptr_t)lds_addr)[k] = ((const uint32_t*)gaddr)[k];
#endif
}

__device__ __forceinline__ void async_wait0() {
#if defined(__gfx1250__)
#if __has_builtin(__builtin_amdgcn_s_wait_asynccnt)
  __builtin_amdgcn_s_wait_asynccnt(0);
#else
  asm volatile("s_wait_asynccnt 0" ::: "memory");
#endif
#endif
}

// copy ndw dwords (ndw % 4 == 0, 16B-aligned both sides) global -> LDS cooperatively
__device__ __forceinline__ void async_copy_16B(void* lds_dst, const void* gsrc,
                                               int ndw, int tid, int nthr) {
  const uint32_t* g = (const uint32_t*)gsrc;
  uint32_t* l = (uint32_t*)lds_dst;
  int nchunk = ndw >> 2;
  for (int i = tid; i < nchunk; i += nthr)
    async_ld_b128((uint32_t)(uintptr_t)(l + 4 * i), g + 4 * i);
}

// ---------------- geometry helpers (shared by both passes for bit-identity) ----------------
struct BEV { float x1, y1, x2, y2; };
struct Anchor { float x, y, z, w, l, h, rot; int j; int r; };

__device__ __forceinline__ float limit_period(float v, float off, float period) {
  return v - floorf(v / period + off) * period;
}

__device__ __forceinline__ BEV make_bev(float x, float y, float wb, float lb) {
  BEV o;
  o.x1 = x - 0.5f * wb; o.y1 = y - 0.5f * lb;
  o.x2 = x + 0.5f * wb; o.y2 = y + 0.5f * lb;
  return o;
}

// general (runtime-theta) nearest-BEV, used for GT boxes only (pass 0, 160 threads)
__device__ __forceinline__ BEV box_bev(float x, float y, float w, float l, float th) {
  float ang = limit_period(th, 0.5f, PI_F);
  bool swap = fabsf(ang) > 0.25f * PI_F;
  return make_bev(x, y, swap ? l : w, swap ? w : l);
}

// Hot-loop IoU: one v_rcp_f32 instead of a 12-op precise divide chain.
// inter==0 -> 0*finite == +0 == 0/u, so the zero case needs no special handling.
// Identical inlined code in pass 1 and pass 2 => bit-identical, so the
// (iou == gt_max) equality test is exact.
__device__ __forceinline__ float iou_fast(float gx1, float gy1, float gx2, float gy2,
                                          float garea, const BEV& ab, float aarea) {
  float iw = fminf(gx2, ab.x2) - fmaxf(gx1, ab.x1);
  float ih = fminf(gy2, ab.y2) - fmaxf(gy1, ab.y1);
  iw = fmaxf(iw, 0.0f); ih = fmaxf(ih, 0.0f);
  float inter = iw * ih;
  float u = fmaxf(garea + aarea - inter, 1e-8f);
#if defined(__gfx1250__)
  return inter * __builtin_amdgcn_rcpf(u);
#else
  return inter / u;
#endif
}

// flat anchor index a = ((y*lx + x)*3 + j)*2 + r
__device__ __forceinline__ Anchor make_anchor(int a, int lx, int ly) {
  Anchor an;
  an.r = a & 1;
  int t = a >> 1;
  an.j = t % NCLS; t /= NCLS;
  int xi = t % lx;
  int yi = t / lx;
  const float* rng = c_rng[an.j];
  const float* sz  = c_sz[an.j];
  float sx = (rng[3] - rng[0]) / (float)lx;
  float sy = (rng[4] - rng[1]) / (float)ly;
  an.x = rng[0] + xi * sx + 0.5f * sx;
  an.y = rng[1] + yi * sy + 0.5f * sy;
  an.z = rng[2] + 0.5f * (rng[5] - rng[2]);
  an.w = sz[0]; an.l = sz[1]; an.h = sz[2];
  an.rot = c_rot[an.r];
  return an;
}

// Anchor rotations are exactly {0.0, 1.57}: swap == (r==1), bit-identical to
// box_bev but with no divide/floor chain.
__device__ __forceinline__ BEV anchor_bev(const Anchor& an) {
  bool swap = (an.r == 1);
  return make_bev(an.x, an.y, swap ? an.l : an.w, swap ? an.w : an.l);
}

// gt_bev layout: 8 floats per box {x1,y1,x2,y2,area,0,0,0} (16B-aligned stride)
#define BEV_STRIDE 8

// ---------------- pass 0: GT -> BEV (+area), zero gt_max accumulators ----------------
__global__ void pp_prep_kernel(const float* __restrict__ gtb,
                               float* __restrict__ gt_bev,
                               uint32_t* __restrict__ gtmax, int B) {
  int t = blockIdx.x * blockDim.x + threadIdx.x;
  if (t < B * G_GT) {
    const float* p = gtb + (size_t)t * 7;
    BEV v = box_bev(p[0], p[1], p[3], p[4], p[6]);
    float* o = gt_bev + (size_t)t * BEV_STRIDE;
    o[0] = v.x1; o[1] = v.y1; o[2] = v.x2; o[3] = v.y2;
    o[4] = (v.x2 - v.x1) * (v.y2 - v.y1);
    o[5] = 0.0f; o[6] = 0.0f; o[7] = 0.0f;
  }
  if (t < B * NCLS * G_GT) gtmax[t] = 0u;   // IoU >= 0, so 0-bits is the identity for max
}

// ---------------- pass 1: gt_max[b][j][g] = max_a IoU ----------------
__global__ void pp_gtmax_kernel(const float* __restrict__ gt_bev,
                                uint32_t* __restrict__ gtmax,
                                const int* __restrict__ p_ly,
                                const int* __restrict__ p_lx, int A) {
  __shared__ __align__(16) float    s_bev[G_GT * BEV_STRIDE];
  __shared__ uint32_t s_max[NCLS * G_GT];
  const int b = blockIdx.y, tid = threadIdx.x, nthr = blockDim.x;

  async_copy_16B(s_bev, gt_bev + (size_t)b * G_GT * BEV_STRIDE, G_GT * BEV_STRIDE, tid, nthr);
  for (int i = tid; i < NCLS * G_GT; i += nthr) s_max[i] = 0u;
  async_wait0();
  __syncthreads();

  const int lx = *p_lx, ly = *p_ly;
  int a = blockIdx.x * blockDim.x + tid;
  if (a < A) {
    Anchor an = make_anchor(a, lx, ly);
    BEV ab = anchor_bev(an);
    float aarea = (ab.x2 - ab.x1) * (ab.y2 - ab.y1);
    #pragma unroll
    for (int g = 0; g < G_GT; ++g) {
      const float* gb = &s_bev[g * BEV_STRIDE];
      float iou = iou_fast(gb[0], gb[1], gb[2], gb[3], gb[4], ab, aarea);
      if (iou > 0.0f)   // most anchors don't overlap a given GT -> few ds atomics
        atomicMax(&s_max[an.j * G_GT + g], __float_as_uint(iou));
    }
  }
  __syncthreads();
  for (int i = tid; i < NCLS * G_GT; i += nthr)
    if (s_max[i] != 0u)
      atomicMax(&gtmax[(size_t)b * NCLS * G_GT + i], s_max[i]);
}

// ---------------- pass 2: per-anchor assignment + outputs ----------------
__global__ void pp_assign_kernel(const float* __restrict__ gt_boxes,
                                 const int* __restrict__ gt_labels,
                                 const float* __restrict__ gt_bev,
                                 const uint32_t* __restrict__ gtmax,
                                 const int* __restrict__ p_ly,
                                 const int* __restrict__ p_lx,
                                 float* __restrict__ out, int A, int B) {
  __shared__ __align__(16) float    s_bev[G_GT * BEV_STRIDE];
  __shared__ __align__(16) float    s_box[G_GT * 7];
  __shared__ __align__(16) int      s_lab[G_GT];
  __shared__ __align__(16) uint32_t s_gmax[NCLS * G_GT];
  const int b = blockIdx.y, tid = threadIdx.x, nthr = blockDim.x;

  async_copy_16B(s_bev,  gt_bev    + (size_t)b * G_GT * BEV_STRIDE, G_GT * BEV_STRIDE, tid, nthr);
  async_copy_16B(s_box,  gt_boxes  + (size_t)b * G_GT * 7,          G_GT * 7,          tid, nthr);
  async_copy_16B(s_lab,  gt_labels + (size_t)b * G_GT,              G_GT,              tid, nthr);
  async_copy_16B(s_gmax, gtmax     + (size_t)b * NCLS * G_GT,       NCLS * G_GT,       tid, nthr);
  async_wait0();
  __syncthreads();

  const int lx = *p_lx, ly = *p_ly;
  int a = blockIdx.x * blockDim.x + tid;
  if (a >= A) return;

  Anchor an = make_anchor(a, lx, ly);
  BEV ab = anchor_bev(an);
  float aarea = (ab.x2 - ab.x1) * (ab.y2 - ab.y1);
  const float thr_neg = c_neg[an.j];   // == min thr
  const float thr_pos = c_pos[an.j];

  float max_ov = -1.0f;
  int max_idx = 0;        // argmax over g, first-max wins (jnp.argmax semantics)
  int forced = -1;        // last g satisfying (iou == gt_max && gt_max >= min) wins
  #pragma unroll
  for (int g = 0; g < G_GT; ++g) {
    const float* gb = &s_bev[g * BEV_STRIDE];
    float iou = iou_fast(gb[0], gb[1], gb[2], gb[3], gb[4], ab, aarea);
    if (iou > max_ov) { max_ov = iou; max_idx = g; }
    float gm = __uint_as_float(s_gmax[an.j * G_GT + g]);
    if (iou == gm && gm >= thr_neg) forced = g;
  }

  int assigned = -1;
  if (max_ov <  thr_neg) assigned = 0;
  if (max_ov >= thr_pos) assigned = max_idx + 1;
  if (forced >= 0)       assigned = forced + 1;

  bool pos = assigned > 0;
  bool negf = assigned == 0;
  int idx = assigned - 1; if (idx < 0) idx = 0;

  float label   = pos ? (float)s_lab[idx] : (float)NCLS;
  float label_w = (pos || negf) ? 1.0f : 0.0f;
  float reg_w   = pos ? 1.0f : 0.0f;

  // precise math for the continuous regression outputs (once per anchor)
  const float* g7 = &s_box[idx * 7];
  float da  = sqrtf(an.w * an.w + an.l * an.l);
  float dx  = (g7[0] - an.x) / da;
  float dy  = (g7[1] - an.y) / da;
  float dz  = ((g7[2] + 0.5f * g7[5]) - (an.z + 0.5f * an.h)) / an.h;
  float dw  = logf(g7[3] / an.w);
  float dl  = logf(g7[4] / an.l);
  float dh  = logf(g7[5] / an.h);
  float dth = g7[6] - an.rot;
  float reg[7] = {dx, dy, dz, dw, dl, dh, dth};

  int dir_t = (int)floorf(limit_period(g7[6], 0.0f, 2.0f * PI_F) / PI_F);
  dir_t = min(max(dir_t, 0), 1);
  float dir_lab = pos ? (float)dir_t : 0.0f;

  // output tuple flattened: labels | label_w | reg | reg_w | dirs | dir_w
  size_t BA   = (size_t)B * (size_t)A;
  size_t base = (size_t)b * (size_t)A + (size_t)a;
  out[base]            = label;
  out[BA + base]       = label_w;
  float* regp = out + 2 * BA + base * 7;
  #pragma unroll
  for (int k = 0; k < 7; ++k) regp[k] = pos ? reg[k] : 0.0f;
  out[9  * BA + base]  = reg_w;
  out[10 * BA + base]  = dir_lab;
  out[11 * BA + base]  = reg_w;
}

extern "C" void kernel_launch(void* const* d_in, const int* in_sizes, int n_in,
                              void* d_out, int out_size, void* d_ws, size_t ws_size,
                              hipStream_t stream) {
  const float* gt_boxes  = (const float*)d_in[0];
  const int*   gt_labels = (const int*)d_in[1];
  const int*   p_ly      = (const int*)d_in[2];
  const int*   p_lx      = (const int*)d_in[3];

  const int B = in_sizes[1] / G_GT;                       // in_sizes[1] == B*G, G fixed at 20
  const int A = (int)((long long)out_size / (12LL * B));  // ly*lx*NCLS*2 per batch

  float*    gt_bev = (float*)d_ws;                                                // B*G*8 floats
  uint32_t* gtmax  = (uint32_t*)((char*)d_ws +
                                 sizeof(float) * (size_t)B * G_GT * BEV_STRIDE);  // B*3*G u32

  {
    int tot = B * NCLS * G_GT;
    pp_prep_kernel<<<(tot + 255) / 256, 256, 0, stream>>>(gt_boxes, gt_bev, gtmax, B);
  }
  int nblk = (A + 255) / 256;
  pp_gtmax_kernel<<<dim3(nblk, B), 256, 0, stream>>>(gt_bev, gtmax, p_ly, p_lx, A);
  pp_assign_kernel<<<dim3(nblk, B), 256, 0, stream>>>(gt_boxes, gt_labels, gt_bev, gtmax,
                                                      p_ly, p_lx, (float*)d_out, A, B);
}